// ChainLVeG_9698036154935
// MI455X (gfx1250) — compile-verified
//
#include <hip/hip_runtime.h>
#include <math.h>

typedef float v2f __attribute__((ext_vector_type(2)));
typedef float v8f __attribute__((ext_vector_type(8)));

#define NTOT 38318      // 46 + 92 + 92 + 4232 + 4*8464
#define DK   512
#define NLBL 46
#define LOG_2PI 1.8378770664093453f

// column offsets of each projection inside the concatenated GEMM output Y[128, NTOT]
#define OFF_WS   0      // NL              (raw)
#define OFF_MS   46     // NL*G            (clip)
#define OFF_VS   138    // NL*G            (clip)
#define OFF_TW   230    // NL*NL*T         (raw)
#define OFF_TPM  4462   // NL*NL*T*G       (clip)
#define OFF_TCM  12926  // NL*NL*T*G       (clip)
#define OFF_TPV  21390  // NL*NL*T*G       (exp(2*clip))  -- only ever used as variance-exp
#define OFF_TCV  29854  // NL*NL*T*G       (clip)

__device__ __forceinline__ float clip1(float v) {
    return fminf(fmaxf(v, -1.0f), 1.0f);
}

__device__ __forceinline__ void map_col(
    int c,
    const float* ws_w, const float* ms_w, const float* vs_w, const float* tw_w,
    const float* tpm_w, const float* tcm_w, const float* tpv_w, const float* tcv_w,
    const float* ws_b, const float* ms_b, const float* vs_b, const float* tw_b,
    const float* tpm_b, const float* tcm_b, const float* tpv_b, const float* tcv_b,
    const float** wp, const float** bp, int* lc, int* nw, int* kind)
{
    if (c < OFF_MS)       { *wp = ws_w;  *bp = ws_b;  *lc = c - OFF_WS;  *nw = 46;   *kind = 0; }
    else if (c < OFF_VS)  { *wp = ms_w;  *bp = ms_b;  *lc = c - OFF_MS;  *nw = 92;   *kind = 1; }
    else if (c < OFF_TW)  { *wp = vs_w;  *bp = vs_b;  *lc = c - OFF_VS;  *nw = 92;   *kind = 1; }
    else if (c < OFF_TPM) { *wp = tw_w;  *bp = tw_b;  *lc = c - OFF_TW;  *nw = 4232; *kind = 0; }
    else if (c < OFF_TCM) { *wp = tpm_w; *bp = tpm_b; *lc = c - OFF_TPM; *nw = 8464; *kind = 1; }
    else if (c < OFF_TPV) { *wp = tcm_w; *bp = tcm_b; *lc = c - OFF_TCM; *nw = 8464; *kind = 1; }
    else if (c < OFF_TCV) { *wp = tpv_w; *bp = tpv_b; *lc = c - OFF_TPV; *nw = 8464; *kind = 2; }
    else                  { *wp = tcv_w; *bp = tcv_b; *lc = c - OFF_TCV; *nw = 8464; *kind = 1; }
}

// -------------------------------------------------------------------------
// Stage 1: fused 8-way projection GEMM, Y[128, NTOT] = x[128,512] @ Wcat + b
// One block = one 16-column tile. W tile (512x16, 32KB) staged to LDS once,
// shared by 8 waves; each wave owns one 16-row tile and drives
// v_wmma_f32_16x16x4_f32 over the K=512 reduction. Weights stream from HBM
// exactly once (bandwidth-roofline for this stage).
// -------------------------------------------------------------------------
__global__ __launch_bounds__(256) void proj_gemm_kernel(
    const float* __restrict__ x,
    const float* ws_w, const float* ms_w, const float* vs_w, const float* tw_w,
    const float* tpm_w, const float* tcm_w, const float* tpv_w, const float* tcv_w,
    const float* ws_b, const float* ms_b, const float* vs_b, const float* tw_b,
    const float* tpm_b, const float* tcm_b, const float* tpv_b, const float* tcv_b,
    float* __restrict__ Y)
{
    __shared__ float Wt[DK * 16];              // W[k][c16] for this column tile
    const int tid = threadIdx.x;
    const int c0  = blockIdx.x * 16;

    // ---- cooperative stage of the 512x16 weight slab into LDS ----
    {
        int c16 = tid & 15;
        int kb  = tid >> 4;                    // 0..15
        int c   = c0 + c16;
        const float* wp; const float* bp; int lc, nw, kind;
        map_col(c < NTOT ? c : (NTOT - 1),
                ws_w, ms_w, vs_w, tw_w, tpm_w, tcm_w, tpv_w, tcv_w,
                ws_b, ms_b, vs_b, tw_b, tpm_b, tcm_b, tpv_b, tcv_b,
                &wp, &bp, &lc, &nw, &kind);
        bool ok = (c < NTOT);
        for (int kk = kb; kk < DK; kk += 16)
            Wt[kk * 16 + c16] = ok ? wp[kk * nw + lc] : 0.0f;
    }
    __syncthreads();

    const int wave = tid >> 5;
    const int lane = tid & 31;
    const int n16  = lane & 15;
    const int ko   = (lane >= 16) ? 2 : 0;     // K split across lane halves
    const int m    = (wave << 4) + n16;        // global row for A loads
    const float* xr = x + (size_t)m * DK;

    v8f acc = {0.f, 0.f, 0.f, 0.f, 0.f, 0.f, 0.f, 0.f};
    #pragma unroll 8
    for (int ks = 0; ks < DK; ks += 4) {
        // A 16x4: lane L holds row M=L&15; VGPR v -> K = ks + ko + v
        v2f a = *(const v2f*)(xr + ks + ko);
        // B 4x16: VGPR v -> row K = ks + ko + v, col = lane&15
        v2f b;
        b.x = Wt[(ks + ko + 0) * 16 + n16];
        b.y = Wt[(ks + ko + 1) * 16 + n16];
        acc = __builtin_amdgcn_wmma_f32_16x16x4_f32(
                  false, a, false, b, (short)0, acc, false, false);
    }

    // ---- epilogue: bias, clip / exp(2*clip), store ----
    int c = c0 + n16;
    const float* wp; const float* bp; int lc, nw, kind;
    map_col(c < NTOT ? c : (NTOT - 1),
            ws_w, ms_w, vs_w, tw_w, tpm_w, tcm_w, tpv_w, tcv_w,
            ws_b, ms_b, vs_b, tw_b, tpm_b, tcm_b, tpv_b, tcv_b,
            &wp, &bp, &lc, &nw, &kind);
    (void)nw;
    if (c < NTOT) {
        float bias = bp[lc];
        int rbase = (wave << 4) + ((lane >= 16) ? 8 : 0);   // C/D layout: VGPR i -> rows i, i+8
        #pragma unroll
        for (int i = 0; i < 8; i++) {
            float v = acc[i] + bias;
            if (kind == 1)      v = clip1(v);
            else if (kind == 2) v = __expf(2.0f * clip1(v));
            Y[(size_t)(rbase + i) * NTOT + c] = v;
        }
    }
}

// -------------------------------------------------------------------------
// Stage 2: child-state Gaussian product.
// cs[b,l,a,j,t] from s[j] x t_c[a,j,t]. Stored as [b*l][j][a*2+t](,g) so
// stage 3 reads a contiguous 92/184-float slab per (b,l,j).
// Stores exp(2*cs_var) instead of cs_var (that's all stage 3 needs).
// -------------------------------------------------------------------------
__global__ __launch_bounds__(256) void cs_kernel(
    const float* __restrict__ Y,
    float* __restrict__ cs_scale,   // [128][46][92]
    float* __restrict__ cs_mu,      // [128][46][92][2]
    float* __restrict__ cs_e1)      // [128][46][92][2]
{
    const int bl = blockIdx.x;                     // b*64 + l
    const float* y = Y + (size_t)bl * NTOT;

    for (int e = threadIdx.x; e < NLBL * NLBL * 2; e += 256) {
        int t  = e & 1;
        int ij = e >> 1;
        int j  = ij % NLBL;
        int a  = ij / NLBL;

        float ssum = y[OFF_WS + j];                // + s_weight
        float muo[2], e1o[2];
        #pragma unroll
        for (int g = 0; g < 2; g++) {
            float m1 = y[OFF_MS  + j * 2 + g];
            float v1 = y[OFF_VS  + j * 2 + g];
            float m2 = y[OFF_TCM + e * 2 + g];     // e == (a*46+j)*2+t == tcm (i,j,t) order
            float v2 = y[OFF_TCV + e * 2 + g];
            float v1s = __expf(2.0f * v1);
            float v2s = __expf(2.0f * v2);
            float add = v1s + v2s;
            float r   = 1.0f / add;
            float la  = __logf(add);
            float d   = m1 - m2;
            ssum += -0.5f * (LOG_2PI + la + d * d * r);
            muo[g] = (m1 * v2s + m2 * v1s) * r;
            float var = v1 + v2 - 0.5f * la;
            e1o[g] = __expf(2.0f * var);
        }
        size_t sbase = ((size_t)bl * NLBL + j) * 92 + (a * 2 + t);
        cs_scale[sbase] = ssum;
        cs_mu[sbase * 2 + 0] = muo[0];
        cs_mu[sbase * 2 + 1] = muo[1];
        cs_e1[sbase * 2 + 0] = e1o[0];
        cs_e1[sbase * 2 + 1] = e1o[1];
    }
}

// -------------------------------------------------------------------------
// Stage 3: the 199MB output. Block = (j, pos, b); cs side (a,t)=92 combos
// and tp side (p,tp)=92 combos staged in LDS, 256 threads emit the 92x92
// tile. All exps pre-hoisted: per (element,g) cost = 1 log + 1 div.
// Store-bandwidth bound by design.
// -------------------------------------------------------------------------
__global__ __launch_bounds__(256) void out_kernel(
    const float* __restrict__ Y,
    const float* __restrict__ cs_scale,
    const float* __restrict__ cs_mu,
    const float* __restrict__ cs_e1,
    const float* __restrict__ mask,
    float* __restrict__ out)
{
    __shared__ float s_sc[92], s_mu[184], s_e1[184];
    __shared__ float s_tw[92], s_tm[184], s_te[184];

    const int j   = blockIdx.x;
    const int pos = blockIdx.y;
    const int b   = blockIdx.z;
    const int bl  = b * 64 + pos;
    const int tid = threadIdx.x;

    const float m0 = mask[bl];
    const size_t cbase = ((size_t)bl * NLBL + j) * 92;
    if (tid < 92)  s_sc[tid] = cs_scale[cbase + tid];
    if (tid < 184) { s_mu[tid] = cs_mu[cbase * 2 + tid]; s_e1[tid] = cs_e1[cbase * 2 + tid]; }

    if (pos < 63) {
        const float* yn = Y + (size_t)(bl + 1) * NTOT;     // trans params at pos+1
        if (tid < 184) {
            s_tm[tid] = yn[OFF_TPM + j * 184 + tid];       // clipped t_p_mu (p,tp,g)
            s_te[tid] = yn[OFF_TPV + j * 184 + tid];       // exp(2*clip(t_p_var))
        }
        if (tid < 92) s_tw[tid] = yn[OFF_TW + j * 92 + tid];
    }
    __syncthreads();

    const size_t obase = (size_t)bl * 389344 + (size_t)j * 184;

    if (pos < 63) {
        const float m1k = mask[bl + 1];
        for (int e = tid; e < 8464; e += 256) {
            int a     = e / 184;
            int inner = e - a * 184;                       // p*4 + t*2 + tp
            int t     = (inner >> 1) & 1;
            int ptp   = ((inner >> 2) << 1) | (inner & 1); // p*2 + tp
            int csi   = a * 2 + t;

            float acc = 0.0f;
            #pragma unroll
            for (int g = 0; g < 2; g++) {
                float m1 = s_mu[csi * 2 + g];
                float e1 = s_e1[csi * 2 + g];
                float m2 = s_tm[ptp * 2 + g];
                float e2 = s_te[ptp * 2 + g];
                float add = e1 + e2;
                float d   = m1 - m2;
                acc += LOG_2PI + __logf(add) + d * d / add;
            }
            float val = (-0.5f * acc + s_tw[ptp]) * m1k + s_sc[csi];
            out[obase + (size_t)a * 8464 + inner] = val * m0;
        }
    } else {
        // last position: cs_scale broadcast over (p,tp)
        for (int e = tid; e < 8464; e += 256) {
            int a     = e / 184;
            int inner = e - a * 184;
            int t     = (inner >> 1) & 1;
            out[obase + (size_t)a * 8464 + inner] = s_sc[a * 2 + t] * m0;
        }
    }
}

extern "C" void kernel_launch(void* const* d_in, const int* in_sizes, int n_in,
                              void* d_out, int out_size, void* d_ws, size_t ws_size,
                              hipStream_t stream)
{
    (void)in_sizes; (void)n_in; (void)out_size; (void)ws_size;

    const float* x     = (const float*)d_in[0];
    const float* mask  = (const float*)d_in[1];
    const float* ws_w  = (const float*)d_in[2];
    const float* ws_b  = (const float*)d_in[3];
    const float* ms_w  = (const float*)d_in[4];
    const float* ms_b  = (const float*)d_in[5];
    const float* vs_w  = (const float*)d_in[6];
    const float* vs_b  = (const float*)d_in[7];
    const float* tw_w  = (const float*)d_in[8];
    const float* tw_b  = (const float*)d_in[9];
    const float* tpm_w = (const float*)d_in[10];
    const float* tpm_b = (const float*)d_in[11];
    const float* tcm_w = (const float*)d_in[12];
    const float* tcm_b = (const float*)d_in[13];
    const float* tpv_w = (const float*)d_in[14];
    const float* tpv_b = (const float*)d_in[15];
    const float* tcv_w = (const float*)d_in[16];
    const float* tcv_b = (const float*)d_in[17];

    // scratch layout (floats): Y | cs_scale | cs_mu | cs_e1  (~30.5 MB total)
    float* Y   = (float*)d_ws;
    float* csS = Y   + (size_t)128 * NTOT;      // 4,904,704
    float* csM = csS + (size_t)128 * 4232;      // 541,696
    float* csE = csM + (size_t)128 * 8464;      // 1,083,392

    proj_gemm_kernel<<<(NTOT + 15) / 16, 256, 0, stream>>>(
        x, ws_w, ms_w, vs_w, tw_w, tpm_w, tcm_w, tpv_w, tcv_w,
        ws_b, ms_b, vs_b, tw_b, tpm_b, tcm_b, tpv_b, tcv_b, Y);

    cs_kernel<<<128, 256, 0, stream>>>(Y, csS, csM, csE);

    out_kernel<<<dim3(NLBL, 64, 2), 256, 0, stream>>>(
        Y, csS, csM, csE, mask, (float*)d_out);
}